// LearnableFilter_76982993814367
// MI455X (gfx1250) — compile-verified
//
#include <hip/hip_runtime.h>

typedef float v2f __attribute__((ext_vector_type(2)));
typedef float v8f __attribute__((ext_vector_type(8)));
typedef int   v4i __attribute__((ext_vector_type(4)));

#define T_LEN   4096
#define NBATCH  8192
#define LSTRIDE 68            // 64 cols + 4 pad floats; LDS row base bank = 4*row (mod 64)
#define WPB     2             // waves per block
#define NTILES  (T_LEN / 64)  // 64 time-columns staged per LDS tile

#if defined(__has_builtin)
#  if __has_builtin(__builtin_amdgcn_global_load_async_to_lds_b128)
#    define HAVE_ASYNC 1
#  endif
#endif

#if defined(HAVE_ASYNC)
typedef __attribute__((address_space(1))) v4i g_v4i;   // global int4
typedef __attribute__((address_space(3))) v4i l_v4i;   // LDS int4
#  if __has_builtin(__builtin_amdgcn_s_wait_asynccnt)
#    define WAIT_ASYNC(n) __builtin_amdgcn_s_wait_asynccnt(n)
#  else
#    define WAIT_ASYNC(n) asm volatile("s_wait_asynccnt %0" :: "i"(n) : "memory")
#  endif
#endif

__global__ __launch_bounds__(32 * WPB)
void LearnableFilter_scan_wmma(const float* __restrict__ x,
                               const float* __restrict__ beta,
                               const float* __restrict__ coupling,
                               const float* __restrict__ initm,
                               float* __restrict__ out)
{
    __shared__ float xs[WPB][2][16 * LSTRIDE];   // double-buffered per wave

    const int lane = threadIdx.x & 31;
    const int wid  = threadIdx.x >> 5;
    const int hi   = (lane >> 4) & 1;
    const int r    = lane & 15;                   // row (A) / time-col j (B,C,D)
    const int rb   = (blockIdx.x * WPB + wid) << 4;

    // fb = sigmoid(beta) * coupling
    const float fb = coupling[0] / (1.0f + __expf(-beta[0]));
    const float om = 1.0f - fb;
    const float f2 = fb * fb, f4 = f2 * f2, f8 = f4 * f4;
    const float a16 = f8 * f8;                    // fb^16 (chunk carry decay)

    // apw = fb^r  (j-th column scale of the rank-1 carry term)
    float apw = 1.0f;
    for (int i = 0; i < r; ++i) apw *= fb;

    // B operand of x-WMMAs: B_t[k][j] = (k<j) ? om*fb^(j-1-k) : 0 ; k = 4t+2hi+u, j=r
    // Dot-product consts for the serial carry: cd[t][u] = om*fb^(15-k)
    v2f bw[4], cd[4];
    #pragma unroll
    for (int t = 0; t < 4; ++t) {
        #pragma unroll
        for (int u = 0; u < 2; ++u) {
            const int k = 4 * t + 2 * hi + u;
            float w = 0.0f;
            if (k < r) { w = om; for (int e = 0; e < r - 1 - k; ++e) w *= fb; }
            bw[t][u] = w;
            float c = om;
            for (int e = 0; e < 15 - k; ++e) c *= fb;
            cd[t][u] = c;
        }
    }

    // B operand of the carry-WMMA: only k=0 row is live (lower half); upper-half slots are 0.
    const v2f bcv = { hi ? 0.0f : apw, 0.0f };

    // per-lane carry state m_{16c}[row = lane&15], replicated across half-waves
    float cpl = initm[rb + r];

    float* xt0 = xs[wid][0];
    float* xt1 = xs[wid][1];

#if defined(HAVE_ASYNC)
    // -------- async global -> LDS pipeline (ASYNCcnt), double buffered --------
    #pragma unroll
    for (int i = 0; i < 8; ++i) {
        const int flat = i * 32 + lane;
        const int rr = flat >> 4, tq = flat & 15;
        const float* gp = x + (size_t)(rb + rr) * T_LEN + tq * 4;
        __builtin_amdgcn_global_load_async_to_lds_b128(
            (g_v4i*)gp, (l_v4i*)&xt0[rr * LSTRIDE + tq * 4], 0, 0);
    }
#else
    float4 nx[8];
    #pragma unroll
    for (int i = 0; i < 8; ++i) {
        const int flat = i * 32 + lane;
        const int rr = flat >> 4, tq = flat & 15;
        nx[i] = *(const float4*)(x + (size_t)(rb + rr) * T_LEN + tq * 4);
    }
#endif

    for (int tile = 0; tile < NTILES; ++tile) {
        const int ct = tile * 64;
        float* xt;

#if defined(HAVE_ASYNC)
        xt = (tile & 1) ? xt1 : xt0;
        if (tile + 1 < NTILES) {
            float* xn = (tile & 1) ? xt0 : xt1;
            #pragma unroll
            for (int i = 0; i < 8; ++i) {
                const int flat = i * 32 + lane;
                const int rr = flat >> 4, tq = flat & 15;
                const float* gp = x + (size_t)(rb + rr) * T_LEN + (ct + 64) + tq * 4;
                __builtin_amdgcn_global_load_async_to_lds_b128(
                    (g_v4i*)gp, (l_v4i*)&xn[rr * LSTRIDE + tq * 4], 0, 0);
            }
            WAIT_ASYNC(8);   // loads retire in order -> current tile's 8 are done
        } else {
            WAIT_ASYNC(0);
        }
#else
        xt = xt0;
        #pragma unroll
        for (int i = 0; i < 8; ++i) {
            const int flat = i * 32 + lane;
            const int rr = flat >> 4, tq = flat & 15;
            *(float4*)&xt[rr * LSTRIDE + tq * 4] = nx[i];
        }
        if (tile + 1 < NTILES) {
            #pragma unroll
            for (int i = 0; i < 8; ++i) {
                const int flat = i * 32 + lane;
                const int rr = flat >> 4, tq = flat & 15;
                nx[i] = *(const float4*)(x + (size_t)(rb + rr) * T_LEN + (ct + 64) + tq * 4);
            }
        }
#endif

        #pragma unroll
        for (int cl = 0; cl < 4; ++cl) {
            const int kb0 = cl * 16;
            const int abase = r * LSTRIDE + kb0 + 2 * hi;
            v2f a0 = *(const v2f*)&xt[abase + 0];
            v2f a1 = *(const v2f*)&xt[abase + 4];
            v2f a2 = *(const v2f*)&xt[abase + 8];
            v2f a3 = *(const v2f*)&xt[abase + 12];

            const float m_cur = cpl;              // m_{16c}

            // ---- off-critical-path D = carry-rank1 + X*W^T (5 WMMAs) ----
            const v2f acv = { m_cur, 0.0f };      // k=0 slot; upper-half slot killed by bcv=0
            v8f acc = {};
            acc = __builtin_amdgcn_wmma_f32_16x16x4_f32(false, acv, false, bcv,   (short)0, acc, false, false);
            acc = __builtin_amdgcn_wmma_f32_16x16x4_f32(false, a3,  false, bw[3], (short)0, acc, false, false);
            acc = __builtin_amdgcn_wmma_f32_16x16x4_f32(false, a2,  false, bw[2], (short)0, acc, false, false);
            acc = __builtin_amdgcn_wmma_f32_16x16x4_f32(false, a1,  false, bw[1], (short)0, acc, false, false);
            acc = __builtin_amdgcn_wmma_f32_16x16x4_f32(false, a0,  false, bw[0], (short)0, acc, false, false);

            // ---- serial carry: m' = fb^16*m + om*sum fb^(15-i)*x_i  (1 FMA on the chain) ----
            float p = cd[0][0] * a0[0];
            p = fmaf(cd[0][1], a0[1], p);
            p = fmaf(cd[1][0], a1[0], p);
            p = fmaf(cd[1][1], a1[1], p);
            p = fmaf(cd[2][0], a2[0], p);
            p = fmaf(cd[2][1], a2[1], p);
            p = fmaf(cd[3][0], a3[0], p);
            p = fmaf(cd[3][1], a3[1], p);
            const float s = p + __shfl_xor(p, 16, 32);   // combine k-halves
            cpl = fmaf(a16, m_cur, s);

            // ---- store D: row = rb+v+8*hi, col = ct+kb0+r (64B contiguous segments) ----
            const size_t colbase = (size_t)(ct + kb0 + r);
            #pragma unroll
            for (int v = 0; v < 8; ++v)
                out[(size_t)(rb + v + 8 * hi) * T_LEN + colbase] = acc[v];
        }
    }
}

extern "C" void kernel_launch(void* const* d_in, const int* in_sizes, int n_in,
                              void* d_out, int out_size, void* d_ws, size_t ws_size,
                              hipStream_t stream) {
    const float* x        = (const float*)d_in[0];
    const float* beta     = (const float*)d_in[1];
    const float* coupling = (const float*)d_in[2];
    const float* initm    = (const float*)d_in[3];
    float* out            = (float*)d_out;

    const int nblocks = NBATCH / (16 * WPB);   // 256 blocks x 2 waves = 512 waves
    LearnableFilter_scan_wmma<<<nblocks, 32 * WPB, 0, stream>>>(
        x, beta, coupling, initm, out);
}